// GenerativeROUNDModel_47605417509471
// MI455X (gfx1250) — compile-verified
//
#include <hip/hip_runtime.h>

typedef __attribute__((ext_vector_type(2))) float v2f;
typedef __attribute__((ext_vector_type(4))) float v4f;
typedef __attribute__((ext_vector_type(8))) float v8f;
typedef __attribute__((ext_vector_type(4))) unsigned int u32x4;
typedef __attribute__((ext_vector_type(4))) int i32x4;
typedef __attribute__((ext_vector_type(8))) int i32x8;

#define H_DIM 1024
#define D_IN  512
#define D_OUT 4096
#define B_SZ  32
#define S_LEN 2048
#define M_TOT (B_SZ * S_LEN)   // 65536

// LDS row pitch for staged Wr tiles: 72 dwords so rows 2 apart are 16 banks
// apart (lane0-15 vs lane16-31 fragment reads hit disjoint banks). The TDM
// produces this pitch natively via pad_interval=64 dw, pad_amount=8 dw.
#define LDS_PITCH   72
#define CHUNK_DW    (48 * LDS_PITCH)          // one staged chunk, dwords
#define CHUNK_BYTES (CHUNK_DW * 4)            // 13824 B

// ---- fp32 WMMA fragment loaders (16x16x4 shape) -------------------------
// A 16x4 (MxK) row-major: lanes 0-15 -> M=lane,    K = k0+{0,1}
//                         lanes 16-31 -> M=lane-16, K = k0+{2,3}
__device__ __forceinline__ v2f load_afrag_f32(const float* __restrict__ A,
                                              int lda, int m0, int k0, int lane) {
  const int m  = m0 + (lane & 15);
  const int kb = k0 + ((lane >> 4) << 1);
  const float* p = A + (long)m * lda + kb;
  v2f a;
  a.x = p[0];
  a.y = p[1];
  return a;
}

// B 4x16 (KxN) row-major: VGPR0 holds rows K=k0 (lanes 0-15) / k0+2 (16-31),
//                         VGPR1 holds rows K=k0+1 / k0+3; N = n per lane.
__device__ __forceinline__ v2f load_bfrag_f32(const float* __restrict__ Bm,
                                              int ldb, int k0, int n, int lane) {
  const int kb = k0 + ((lane >> 4) << 1);
  v2f b;
  b.x = Bm[(long)kb * ldb + n];
  b.y = Bm[(long)(kb + 1) * ldb + n];
  return b;
}

// ---- Tensor Data Mover: 16 rows x 64 f32 cols, row stride D_OUT,
//      into LDS at `lds_byte` with pitch 72 dwords (8-dword pad per row).
__device__ __forceinline__ void tdm_load_tile(unsigned lds_byte,
                                              const float* gsrc) {
  const unsigned long long ga = (unsigned long long)(const void*)gsrc;
  u32x4 g0;
  g0[0] = 1u;                                   // count=1, user descriptor
  g0[1] = lds_byte;                             // lds_addr [63:32]
  g0[2] = (unsigned)ga;                         // global_addr low
  g0[3] = (unsigned)(ga >> 32) | (2u << 30);    // global_addr hi | type=2
  i32x8 g1;
  g1[0] = (2 << 16)        // data_size = 4B
        | (1 << 20)        // pad_enable
        | (5 << 22)        // pad_interval: 64 dwords
        | (7 << 25);       // pad_amount: 8 dwords
  g1[1] = (int)(4096u << 16);        // tensor_dim0 = 4096 ([15:0] -> w1[31:16])
  g1[2] = 0;                         // dim0 hi | tensor_dim1 lo (64K -> lo=0)
  g1[3] = (64 << 16) | 1;            // tile_dim0=64 | tensor_dim1 hi (=1)
  g1[4] = 16;                        // tile_dim1=16, tile_dim2=0
  g1[5] = 4096;                      // tensor_dim0_stride low32 = D_OUT
  g1[6] = 0;
  g1[7] = 0;
  const i32x4 z4 = {0, 0, 0, 0};
#if __clang_major__ >= 23
  const i32x8 z8 = {0, 0, 0, 0, 0, 0, 0, 0};
  __builtin_amdgcn_tensor_load_to_lds(g0, g1, z4, z4, z8, 0);
#else
  __builtin_amdgcn_tensor_load_to_lds(g0, g1, z4, z4, 0);
#endif
}

// =========================================================================
// K1: delta[b,s,h] = omega[h] + k[h] * wrap(x[b,s,:]@We[:,h] + be[h])
//     wrap(t) = atan2(sin t, cos t) = t - 2*pi*rint(t/(2*pi))
// Block = 4 waves, block tile 64(M)x64(N), wave tile 16x64.
// =========================================================================
__global__ __launch_bounds__(128) void k1_embed_delta(
    const float* __restrict__ x, const float* __restrict__ We,
    const float* __restrict__ be, const float* __restrict__ omega,
    const float* __restrict__ kc, float* __restrict__ delta_out) {
  const int lane = threadIdx.x & 31;
  const int wave = threadIdx.x >> 5;
  const int m0 = blockIdx.y * 64 + wave * 16;
  const int n0 = blockIdx.x * 64;
  const int nl = n0 + (lane & 15);

  v8f acc[4] = {v8f{}, v8f{}, v8f{}, v8f{}};

  for (int k0 = 0; k0 < D_IN; k0 += 4) {
    const v2f a = load_afrag_f32(x, D_IN, m0, k0, lane);
#pragma unroll
    for (int j = 0; j < 4; ++j) {
      const v2f b = load_bfrag_f32(We, H_DIM, k0, nl + j * 16, lane);
      acc[j] = __builtin_amdgcn_wmma_f32_16x16x4_f32(
          false, a, false, b, (short)0, acc[j], false, false);
    }
  }

  const float TWO_PI     = 6.2831853071795864769f;
  const float INV_TWO_PI = 0.15915494309189533577f;
  const int mrow = m0 + ((lane >> 4) << 3);
#pragma unroll
  for (int j = 0; j < 4; ++j) {
    const int n = nl + j * 16;
    const float bev = be[n];
    const float om  = omega[n];
    const float kk  = kc[n];
#pragma unroll
    for (int v = 0; v < 8; ++v) {
      const float pt = acc[j][v] + bev;
      const float w  = pt - TWO_PI * rintf(pt * INV_TWO_PI);  // == atan2(sin,cos)
      delta_out[(long)(mrow + v) * H_DIM + n] = om + kk * w;
    }
  }
}

// =========================================================================
// K2: in-place cumsum over S for each (b,h) column. Coalesced over h.
// =========================================================================
__global__ __launch_bounds__(256) void k2_scan(float* __restrict__ ph) {
  const int t = blockIdx.x * 256 + threadIdx.x;   // 0 .. B*H-1
  const int b = t >> 10;                          // / H_DIM
  const int h = t & (H_DIM - 1);
  float* p = ph + (long)b * S_LEN * H_DIM + h;
  float acc = 0.0f;
  for (int s = 0; s < S_LEN; s += 4) {
    const float d0 = p[0];
    const float d1 = p[1 * H_DIM];
    const float d2 = p[2 * H_DIM];
    const float d3 = p[3 * H_DIM];
    acc += d0; p[0 * H_DIM] = acc;
    acc += d1; p[1 * H_DIM] = acc;
    acc += d2; p[2 * H_DIM] = acc;
    acc += d3; p[3 * H_DIM] = acc;
    p += 4 * H_DIM;
  }
}

// =========================================================================
// K3: logits = [cos ph | sin ph | ph] @ Wr + br, feats built on the fly.
// Block = 4 waves stacked in M; block tile 128(M) x 64(N).
// Wave tile 32x64: 8 accumulators; each B fragment feeds 2 M-tiles,
// each A fragment feeds 4 N-tiles -> 24 WMMAs per 4-deep K step.
// Wr K-chunks (3 sections x 16 rows x 64 cols) DMA'd into LDS by the
// Tensor Data Mover (wave 0 issues 3 descriptors), double-buffered so
// the next chunk's DMA overlaps the current chunk's 96 WMMAs.
// =========================================================================
__global__ __launch_bounds__(128) void k3_readout(
    const float* __restrict__ ph, const float* __restrict__ Wr,
    const float* __restrict__ br, float* __restrict__ logits) {
  __shared__ float sB[2][CHUNK_DW];   // 2 x 13.8 KB of the 320KB/WGP pool

  const int tid  = threadIdx.x;
  const int lane = tid & 31;
  const int wave = tid >> 5;
  const int m0w  = blockIdx.y * 128 + wave * 32;   // wave's 32-row strip
  const int n0   = blockIdx.x * 64;
  const int lnl  = lane & 15;                      // fragment column
  const int koff = (lane >> 4) << 1;               // 0 or 2: K sub-offset

  // raw LDS byte offset of the staging buffers (generic ptr low 32 bits)
  const unsigned lds0 = (unsigned)(unsigned long long)(void*)&sB[0][0];

  v8f acc[2][4] = {{v8f{}, v8f{}, v8f{}, v8f{}}, {v8f{}, v8f{}, v8f{}, v8f{}}};

  // ---- prologue: DMA chunk 0 into buffer 0 -------------------------------
  if (wave == 0) {
#pragma unroll
    for (int sec = 0; sec < 3; ++sec)
      tdm_load_tile(lds0 + sec * (16 * LDS_PITCH * 4),
                    Wr + (long)(sec * H_DIM) * D_OUT + n0);
    __builtin_amdgcn_s_wait_tensorcnt(0);
  }
  __syncthreads();

  int buf = 0;
  for (int h0 = 0; h0 < H_DIM; h0 += 16) {
    // ---- kick next chunk's DMA into the other buffer ---------------------
    if (wave == 0 && h0 + 16 < H_DIM) {
#pragma unroll
      for (int sec = 0; sec < 3; ++sec)
        tdm_load_tile(lds0 + (buf ^ 1) * CHUNK_BYTES + sec * (16 * LDS_PITCH * 4),
                      Wr + (long)(sec * H_DIM + h0 + 16) * D_OUT + n0);
    }

    // ---- compute from current buffer ------------------------------------
    const float* sbuf = &sB[buf][0];
    for (int hh = 0; hh < 16; hh += 4) {
      // A fragments: load ph, build cos/sin/identity features
      const v2f p0 = load_afrag_f32(ph, H_DIM, m0w,      h0 + hh, lane);
      const v2f p1 = load_afrag_f32(ph, H_DIM, m0w + 16, h0 + hh, lane);
      float s0x, c0x, s0y, c0y, s1x, c1x, s1y, c1y;
      sincosf(p0.x, &s0x, &c0x);  sincosf(p0.y, &s0y, &c0y);
      sincosf(p1.x, &s1x, &c1x);  sincosf(p1.y, &s1y, &c1y);
      v2f a[3][2];
      a[0][0].x = c0x; a[0][0].y = c0y;  a[0][1].x = c1x; a[0][1].y = c1y;  // cos
      a[1][0].x = s0x; a[1][0].y = s0y;  a[1][1].x = s1x; a[1][1].y = s1y;  // sin
      a[2][0] = p0;                      a[2][1] = p1;                      // ph

      // B fragments from LDS; each reused by both M-tiles
#pragma unroll
      for (int sec = 0; sec < 3; ++sec) {
        const float* sRow = sbuf + (sec * 16 + hh + koff) * LDS_PITCH + lnl;
#pragma unroll
        for (int j = 0; j < 4; ++j) {
          v2f b;
          b.x = sRow[j * 16];                 // row kb
          b.y = sRow[j * 16 + LDS_PITCH];     // row kb+1 (ds_load_2addr pair)
          acc[0][j] = __builtin_amdgcn_wmma_f32_16x16x4_f32(
              false, a[sec][0], false, b, (short)0, acc[0][j], false, false);
          acc[1][j] = __builtin_amdgcn_wmma_f32_16x16x4_f32(
              false, a[sec][1], false, b, (short)0, acc[1][j], false, false);
        }
      }
    }

    // publish next buffer: issuing wave drains TENSORcnt, then all sync
    if (wave == 0) __builtin_amdgcn_s_wait_tensorcnt(0);
    __syncthreads();
    buf ^= 1;
  }

  // ---- epilogue: add bias, store 32x64 tile ------------------------------
  const int mrow = m0w + ((lane >> 4) << 3);
#pragma unroll
  for (int mi = 0; mi < 2; ++mi) {
#pragma unroll
    for (int j = 0; j < 4; ++j) {
      const int n = n0 + j * 16 + lnl;
      const float brv = br[n];
#pragma unroll
      for (int v = 0; v < 8; ++v) {
        logits[(long)(mrow + mi * 16 + v) * D_OUT + n] = acc[mi][j][v] + brv;
      }
    }
  }
}

extern "C" void kernel_launch(void* const* d_in, const int* in_sizes, int n_in,
                              void* d_out, int out_size, void* d_ws, size_t ws_size,
                              hipStream_t stream) {
  (void)in_sizes; (void)n_in; (void)d_ws; (void)ws_size; (void)out_size;
  const float* x     = (const float*)d_in[0];
  const float* We    = (const float*)d_in[1];
  const float* be    = (const float*)d_in[2];
  const float* omega = (const float*)d_in[3];
  const float* kcp   = (const float*)d_in[4];
  const float* Wr    = (const float*)d_in[5];
  const float* br    = (const float*)d_in[6];

  float* logits = (float*)d_out;                              // (B,S,D_OUT)
  float* ph     = (float*)d_out + (size_t)M_TOT * D_OUT;      // (B,S,H)

  // K1: delta written into ph region (scanned in place next)
  dim3 g1(H_DIM / 64, M_TOT / 64);          // 16 x 1024
  k1_embed_delta<<<g1, 128, 0, stream>>>(x, We, be, omega, kcp, ph);

  // K2: in-place prefix sum over S
  k2_scan<<<(B_SZ * H_DIM) / 256, 256, 0, stream>>>(ph);

  // K3: readout GEMM, TDM double-buffered Wr staging, on-the-fly features
  dim3 g3(D_OUT / 64, M_TOT / 128);         // 64 x 512
  k3_readout<<<g3, 128, 0, stream>>>(ph, Wr, br, logits);
}